// MySAGEConvBlock_18459769438292
// MI455X (gfx1250) — compile-verified
//
#include <hip/hip_runtime.h>

#define N_NODES 100000
#define N_EDGES 1000000
#define DIM 64
#define BN_EPS 1e-5f
#define WPAD 68  // LDS row stride (floats): (4n + k%4) bank pattern -> conflict-free b64 reads

typedef __attribute__((ext_vector_type(2))) float v2f;
typedef __attribute__((ext_vector_type(8))) float v8f;

// ---------------------------------------------------------------------------
// Kernel 1: fold self-loops into init (summed = x, deg = 1), zero BN stats.
// ---------------------------------------------------------------------------
__global__ void init_kernel(const float* __restrict__ x, float* __restrict__ summed,
                            float* __restrict__ deg, float* __restrict__ stats) {
    size_t i = (size_t)blockIdx.x * blockDim.x + threadIdx.x;
    size_t stride = (size_t)gridDim.x * blockDim.x;
    const float4* x4 = (const float4*)x;
    float4* s4 = (float4*)summed;
    size_t n4 = (size_t)N_NODES * DIM / 4;
    for (size_t j = i; j < n4; j += stride) s4[j] = x4[j];
    for (size_t j = i; j < N_NODES; j += stride) deg[j] = 1.0f;
    if (i < 2 * DIM) stats[i] = 0.0f;
}

// ---------------------------------------------------------------------------
// Kernel 2: edge scatter-add. One wave32 per edge; lane l owns features 2l,2l+1.
// ---------------------------------------------------------------------------
__global__ void edge_scatter_kernel(const float* __restrict__ x, const int* __restrict__ ei,
                                    float* __restrict__ summed, float* __restrict__ deg) {
    int t = blockIdx.x * blockDim.x + threadIdx.x;
    int e = t >> 5;
    int lane = t & 31;
    if (e >= N_EDGES) return;
    int src = ei[e];
    int dst = ei[N_EDGES + e];
    float2 v = ((const float2*)(x + (size_t)src * DIM))[lane];
    float* srow = summed + (size_t)dst * DIM + 2 * lane;
    atomicAdd(srow, v.x);
    atomicAdd(srow + 1, v.y);
    if (lane == 0) atomicAdd(&deg[dst], 1.0f);
}

// ---------------------------------------------------------------------------
// Kernel 3: fused scatter-mean finish + GEMM (fp32 WMMA 16x16x4) + bias
//           + BN statistics accumulation. Each wave -> 16 rows x 64 cols.
//           out[m][n] = sum_k (summed[m][k]/deg[m]) * W[n][k] + b[n]
// ---------------------------------------------------------------------------
__global__ __launch_bounds__(256) void gemm_bn_stats_kernel(
    const float* __restrict__ summed, const float* __restrict__ deg,
    const float* __restrict__ W, const float* __restrict__ bias,
    float* __restrict__ out_pre, float* __restrict__ stats) {
    __shared__ float Wl[DIM * WPAD];
    __shared__ float sum_l[DIM];
    __shared__ float sq_l[DIM];

    int tid = threadIdx.x;
    for (int i = tid; i < DIM * DIM; i += 256) {
        int n = i >> 6, k = i & 63;
        Wl[n * WPAD + k] = W[i];
    }
    if (tid < DIM) { sum_l[tid] = 0.0f; sq_l[tid] = 0.0f; }
    __syncthreads();

    int wave = tid >> 5;
    int lane = tid & 31;
    int lm = lane & 15;   // M (for A) / N (for B) within tile
    int hi = lane >> 4;   // hi half-wave handles K+2
    int row_base = (blockIdx.x * 8 + wave) * 16;

    // A rows: clamp out-of-range rows (EXEC must stay all-1s for WMMA);
    // masked later for stores/stats.
    int arow = row_base + lm;
    if (arow >= N_NODES) arow = N_NODES - 1;
    float rdeg = 1.0f / deg[arow];
    const float* aptr = summed + (size_t)arow * DIM + 2 * hi;

    v8f acc[4] = {};
#pragma unroll
    for (int kk = 0; kk < DIM; kk += 4) {
        // A fragment (16x4 f32): lane lm = row, VGPR v = K = kk + 2*hi + v
        float2 av = *(const float2*)(aptr + kk);
        v2f A;
        A.x = av.x * rdeg;
        A.y = av.y * rdeg;
        // B fragments (4x16 f32): B[k][n] = W[n][k], n = nt*16 + lm
        const float* wb = &Wl[lm * WPAD + kk + 2 * hi];
        float2 t0 = *(const float2*)(wb);
        float2 t1 = *(const float2*)(wb + 16 * WPAD);
        float2 t2 = *(const float2*)(wb + 32 * WPAD);
        float2 t3 = *(const float2*)(wb + 48 * WPAD);
        v2f B0 = {t0.x, t0.y}, B1 = {t1.x, t1.y}, B2 = {t2.x, t2.y}, B3 = {t3.x, t3.y};
        acc[0] = __builtin_amdgcn_wmma_f32_16x16x4_f32(false, A, false, B0, (short)0, acc[0], false, false);
        acc[1] = __builtin_amdgcn_wmma_f32_16x16x4_f32(false, A, false, B1, (short)0, acc[1], false, false);
        acc[2] = __builtin_amdgcn_wmma_f32_16x16x4_f32(false, A, false, B2, (short)0, acc[2], false, false);
        acc[3] = __builtin_amdgcn_wmma_f32_16x16x4_f32(false, A, false, B3, (short)0, acc[3], false, false);
    }

    // Epilogue: bias, store pre-BN activations, accumulate per-feature sums.
    // C/D layout: VGPR r, lane<16 -> row r, lane>=16 -> row 8+r; col = lm.
#pragma unroll
    for (int nt = 0; nt < 4; ++nt) {
        int col = nt * 16 + lm;
        float bv = bias[col];
        float s = 0.0f, s2 = 0.0f;
#pragma unroll
        for (int r = 0; r < 8; ++r) {
            int row = row_base + 8 * hi + r;
            float val = acc[nt][r] + bv;
            if (row < N_NODES) {
                out_pre[(size_t)row * DIM + col] = val;
                s += val;
                s2 += val * val;
            }
        }
        atomicAdd(&sum_l[col], s);
        atomicAdd(&sq_l[col], s2);
    }
    __syncthreads();
    if (tid < DIM) {
        atomicAdd(&stats[tid], sum_l[tid]);
        atomicAdd(&stats[DIM + tid], sq_l[tid]);
    }
}

// ---------------------------------------------------------------------------
// Kernel 4: finalize BN affine params: scale = gamma*rsqrt(var+eps),
//           shift = beta - mean*scale  (biased variance, training mode).
// ---------------------------------------------------------------------------
__global__ void bn_params_kernel(const float* __restrict__ stats,
                                 const float* __restrict__ gamma,
                                 const float* __restrict__ beta,
                                 float* __restrict__ params) {
    int c = threadIdx.x;
    if (c < DIM) {
        const float invN = 1.0f / (float)N_NODES;
        float mean = stats[c] * invN;
        float var = stats[DIM + c] * invN - mean * mean;
        float sc = gamma[c] * rsqrtf(var + BN_EPS);
        params[c] = sc;
        params[DIM + c] = beta[c] - mean * sc;
    }
}

// ---------------------------------------------------------------------------
// Kernel 5: in-place normalize + residual + ReLU over d_out (float4 stream).
// ---------------------------------------------------------------------------
__global__ void finalize_kernel(const float* __restrict__ x,
                                const float* __restrict__ params,
                                float* __restrict__ out) {
    __shared__ float sc[DIM], sh[DIM];
    if (threadIdx.x < DIM) {
        sc[threadIdx.x] = params[threadIdx.x];
        sh[threadIdx.x] = params[DIM + threadIdx.x];
    }
    __syncthreads();
    size_t i = (size_t)blockIdx.x * blockDim.x + threadIdx.x;
    size_t stride = (size_t)gridDim.x * blockDim.x;
    size_t n4 = (size_t)N_NODES * DIM / 4;
    const float4* x4 = (const float4*)x;
    float4* o4 = (float4*)out;
    for (size_t j = i; j < n4; j += stride) {
        float4 o = o4[j];
        float4 xv = x4[j];
        int c0 = (int)((j * 4) & (DIM - 1));
        o.x = fmaxf(o.x * sc[c0 + 0] + sh[c0 + 0] + xv.x, 0.0f);
        o.y = fmaxf(o.y * sc[c0 + 1] + sh[c0 + 1] + xv.y, 0.0f);
        o.z = fmaxf(o.z * sc[c0 + 2] + sh[c0 + 2] + xv.z, 0.0f);
        o.w = fmaxf(o.w * sc[c0 + 3] + sh[c0 + 3] + xv.w, 0.0f);
        o4[j] = o;
    }
}

// ---------------------------------------------------------------------------
extern "C" void kernel_launch(void* const* d_in, const int* in_sizes, int n_in,
                              void* d_out, int out_size, void* d_ws, size_t ws_size,
                              hipStream_t stream) {
    const float* x     = (const float*)d_in[0];
    const int*   ei    = (const int*)d_in[1];   // [2, E] int32 (JAX w/o x64)
    const float* lw    = (const float*)d_in[2]; // [DIM, DIM]
    const float* lb    = (const float*)d_in[3];
    const float* gamma = (const float*)d_in[4];
    const float* beta  = (const float*)d_in[5];
    float* out = (float*)d_out;

    // Workspace layout (~26.0 MB): out_pre lives in d_out (rewritten in-place).
    char* ws = (char*)d_ws;
    float* summed = (float*)(ws);                              // N*DIM*4 = 25,600,000 B
    float* deg    = (float*)(ws + (size_t)N_NODES * DIM * 4);  // N*4     =    400,000 B
    float* stats  = (float*)(ws + (size_t)N_NODES * DIM * 4 + (size_t)N_NODES * 4); // 512 B
    float* params = stats + 2 * DIM;                           // 512 B

    init_kernel<<<2048, 256, 0, stream>>>(x, summed, deg, stats);

    int edge_blocks = (N_EDGES * 32 + 255) / 256;  // one wave32 per edge
    edge_scatter_kernel<<<edge_blocks, 256, 0, stream>>>(x, ei, summed, deg);

    int gemm_blocks = (N_NODES + 127) / 128;       // 8 waves x 16 rows per block
    gemm_bn_stats_kernel<<<gemm_blocks, 256, 0, stream>>>(summed, deg, lw, lb, out, stats);

    bn_params_kernel<<<1, 64, 0, stream>>>(stats, gamma, beta, params);

    finalize_kernel<<<2048, 256, 0, stream>>>(x, params, out);
}